// PhysicsLoss_63350767616155
// MI455X (gfx1250) — compile-verified
//
#include <hip/hip_runtime.h>
#include <hip/hip_bf16.h>
#include <stdint.h>

// ---------- CDNA5 feature probes (compile-time) ----------
#if __has_builtin(__builtin_amdgcn_global_load_async_to_lds_b128)
#define HAVE_ASYNC_LDS 1
#endif
#if __has_builtin(__builtin_amdgcn_wmma_f32_16x16x4_f32)
#define HAVE_WMMA_F32X4 1
#endif

typedef __attribute__((ext_vector_type(2))) float v2f;
typedef __attribute__((ext_vector_type(8))) float v8f;
typedef __attribute__((ext_vector_type(4))) int   v4i;

typedef __attribute__((address_space(1))) v4i as1_v4i;
typedef __attribute__((address_space(3))) v4i as3_v4i;

// Cast generic pointers to specific address spaces via integer round-trip.
// Global: full 64-bit value is the global address.
// LDS: generic LDS pointer keeps the byte offset in addr[31:0] (ISA 10.2).
__device__ __forceinline__ as1_v4i* to_global_v4(const void* p) {
    return (as1_v4i*)(uintptr_t)p;
}
__device__ __forceinline__ as3_v4i* to_lds_v4(const void* p) {
    return (as3_v4i*)(uint32_t)(uintptr_t)p;
}

// ---------- exact 256-element sum by one full wave (lane = 0..31) ----------
// Uses V_WMMA_F32_16X16X4_F32 with B == ones: D[m,n] = sum_k A[m,k] (+C).
// 4 chained WMMAs consume all 256 values exactly once; then sum the 8 D
// registers per lane and add the opposite half-wave (rows 8..15).
__device__ __forceinline__ float wave_sum256(const float* vals, int lane) {
#ifdef HAVE_WMMA_F32X4
    v2f ones; ones.x = 1.0f; ones.y = 1.0f;
    v8f acc = {0.f, 0.f, 0.f, 0.f, 0.f, 0.f, 0.f, 0.f};
    const int m  = lane & 15;
    const int kb = (lane < 16) ? 0 : 2;   // A 16x4 layout: lanes 0-15 k={0,1}, 16-31 k={2,3}
#pragma unroll
    for (int t = 0; t < 4; ++t) {
        v2f a;
        a.x = vals[t * 64 + m * 4 + kb + 0];
        a.y = vals[t * 64 + m * 4 + kb + 1];
        acc = __builtin_amdgcn_wmma_f32_16x16x4_f32(
            /*neg_a=*/false, a, /*neg_b=*/false, ones,
            /*c_mod=*/(short)0, acc, /*reuse_a=*/false, /*reuse_b=*/false);
    }
    float s = ((acc[0] + acc[1]) + (acc[2] + acc[3])) +
              ((acc[4] + acc[5]) + (acc[6] + acc[7]));
    s += __shfl_xor(s, 16, 32);           // add rows 8..15 half
    return s;                             // total replicated across all lanes
#else
    float s = 0.f;
    for (int i = lane; i < 256; i += 32) s += vals[i];
#pragma unroll
    for (int off = 16; off > 0; off >>= 1) s += __shfl_xor(s, off, 32);
    return s;
#endif
}

// ---------- kernel 1: one workgroup per trajectory ----------
// blockDim.x == 256; requires T <= 1024, max_lag <= 256 (reference: 1024/256).
__global__ __launch_bounds__(256)
void msd_loss_kernel(const float* __restrict__ alpha_pred,
                     const float* __restrict__ trajectory,
                     const int*   __restrict__ lengths,
                     float* __restrict__ per_traj,
                     int T) {
    __shared__ float2 s_traj[1024];
    __shared__ float  s_red[256];
    __shared__ float  s_intercept;

    const int tid = threadIdx.x;
    const int b   = blockIdx.x;
    const int P       = T - 1;
    const int max_lag = (T / 4) > 1 ? (T / 4) : 1;

    // ---- stage trajectory[b] (T float2 = 8*T bytes) into LDS (async DMA path) ----
    const char* gbase = (const char*)(trajectory + (size_t)b * (size_t)T * 2);
    const int nbytes = T * 8;
#ifdef HAVE_ASYNC_LDS
    {
        char* lbase = (char*)s_traj;
        for (int off = tid * 16; off < nbytes; off += 256 * 16) {
            __builtin_amdgcn_global_load_async_to_lds_b128(
                to_global_v4(gbase + off), to_lds_v4(lbase + off), 0, 0);
        }
#if __has_builtin(__builtin_amdgcn_s_wait_asynccnt)
        __builtin_amdgcn_s_wait_asynccnt(0);
#else
        asm volatile("s_wait_asynccnt 0x0" ::: "memory");
#endif
    }
#else
    {
        const float4* g4 = (const float4*)gbase;
        float4* s4 = (float4*)s_traj;
        for (int i = tid; i < nbytes / 16; i += 256) s4[i] = g4[i];
    }
#endif
    __syncthreads();

    const float alpha = alpha_pred[b];
    const int   len   = lengths[b];

    // ---- per-lane MSD for lag L = tid+1 ----
    float resid = 0.f, mask = 0.f, log_msd = 0.f, log_lag = 0.f;
    const int L = tid + 1;
    if (L <= max_lag) {
        // valid <=> p < min(len-L, T-L, P); (p+L<T makes end_c clamp irrelevant)
        int NL = len - L;
        int lim = T - L;
        if (NL > lim) NL = lim;
        if (NL > P)   NL = P;
        if (NL < 0)   NL = 0;

        float acc = 0.f;
#pragma unroll 4
        for (int p = 0; p < NL; ++p) {
            float2 e = s_traj[p + L];
            float2 s0 = s_traj[p];
            float dx = e.x - s0.x;
            float dy = e.y - s0.y;
            acc = fmaf(dx, dx, acc);
            acc = fmaf(dy, dy, acc);
        }
        float counts = (float)(NL > 1 ? NL : 1);
        float msd = acc / counts;
        mask    = (len > L) ? 1.f : 0.f;
        log_msd = logf(msd + 1e-8f);
        log_lag = logf((float)L);
        resid   = (log_msd - alpha * log_lag) * mask;
    }

    // denom = max(sum_L [len > L], 1) has closed form: clamp(len-1, 0, max_lag)
    int cnt = len - 1;
    if (cnt < 0) cnt = 0;
    if (cnt > max_lag) cnt = max_lag;
    const float denom = (cnt > 1) ? (float)cnt : 1.0f;

    // ---- reduction 1: intercept = sum(resid*mask)/denom ----
    s_red[tid] = resid;
    __syncthreads();
    if (tid < 32) {                       // full wave 0, EXEC all ones (WMMA requirement)
        float S = wave_sum256(s_red, tid);
        if (tid == 0) s_intercept = S / denom;
    }
    __syncthreads();
    const float intercept = s_intercept;

    // ---- reduction 2: per_traj = sum(sq_err*mask)/denom ----
    float err = alpha * log_lag + intercept - log_msd;
    s_red[tid] = err * err * mask;
    __syncthreads();
    if (tid < 32) {
        float S2 = wave_sum256(s_red, tid);
        if (tid == 0) per_traj[b] = S2 / denom;
    }
}

// ---------- kernel 2: deterministic mean over B values ----------
__global__ __launch_bounds__(32)
void mean_kernel(const float* __restrict__ per_traj, float* __restrict__ out, int B) {
    const int lane = threadIdx.x;
    float v = 0.f;
    for (int i = lane; i < B; i += 32) v += per_traj[i];
#pragma unroll
    for (int off = 16; off > 0; off >>= 1) v += __shfl_down(v, off, 32);
    if (lane == 0) out[0] = v / (float)B;
}

extern "C" void kernel_launch(void* const* d_in, const int* in_sizes, int n_in,
                              void* d_out, int out_size, void* d_ws, size_t ws_size,
                              hipStream_t stream) {
    const float* alpha_pred = (const float*)d_in[0];   // (B,) f32
    const float* trajectory = (const float*)d_in[1];   // (B,T,2) f32
    const int*   lengths    = (const int*)d_in[2];     // (B,) int

    const int B  = in_sizes[0];
    const int TD = in_sizes[1] / B;
    const int T  = TD / 2;

    float* ws = (float*)d_ws;                          // B floats of scratch

    msd_loss_kernel<<<B, 256, 0, stream>>>(alpha_pred, trajectory, lengths, ws, T);
    mean_kernel<<<1, 32, 0, stream>>>(ws, (float*)d_out, B);
}